// ColBERTer_53549652246634
// MI455X (gfx1250) — compile-verified
//
#include <hip/hip_runtime.h>
#include <hip/hip_bf16.h>

// ColBERTer scoring for MI455X (gfx1250), wave32, f32 WMMA 16x16x4.
// B=128, LQ=32, LD=512, H=768, C=128.
//
// Kernel 1 (proj): P = X @ W_comp + b_comp, fused importance/mask epilogue.
//   - one wave per 16-row strip, 8 N-tiles (C=128) of f32 accumulators
//   - W_comp staged in LDS pair-interleaved: sW[p][c] = (W[2p][c], W[2p+1][c])
//     so each WMMA B-fragment is ONE aligned ds_load_b64 (no v_mov assembly);
//     pair-row pitch 144 => halves hit disjoint bank sets
//   - A strip staged in LDS (padded stride 68), fragment = one ds_load_b64
// Kernel 2 (score): per-batch q_vecs @ d_vecs^T with f32 WMMA, masked max,
//   masked sum, cls dot, sigmoid merge.

typedef __attribute__((ext_vector_type(2))) float v2f;
typedef __attribute__((ext_vector_type(8))) float v8f;

#define BCOUNT 128
#define LQ     32
#define LD     512
#define HDIM   768
#define CDIM   128

#define DOC_ROWS   (BCOUNT * LD)     // 65536
#define QRY_ROWS   (BCOUNT * LQ)     // 4096
#define DOC_TILES  (DOC_ROWS / 16)   // 4096
#define QRY_TILES  (QRY_ROWS / 16)   // 256
#define ALL_TILES  (DOC_TILES + QRY_TILES) // 4352

#define KC        64                 // K-chunk staged in LDS (32 pair-rows)
#define NCHUNK    (HDIM / KC)        // 12
#define SWP_PITCH 144                // pairs per pair-row: 2*144 % 64 == 32
#define SA_PITCH  68                 // 64 + 4 pad

__global__ __launch_bounds__(256) void colberter_proj(
    const float* __restrict__ qh,    // (B*LQ, H)
    const float* __restrict__ dh,    // (B*LD, H)
    const int*   __restrict__ qmask, // (B*LQ)
    const int*   __restrict__ dmask, // (B*LD)
    const float* __restrict__ W,     // (H, C)
    const float* __restrict__ bcomp, // (C)
    const float* __restrict__ wstop, // (C)
    const float* __restrict__ bstop, // (1)
    float* __restrict__ qvecs,       // (B*LQ, C)
    float* __restrict__ dvecs)       // (B*LD, C)
{
    __shared__ float sW[(KC / 2) * SWP_PITCH * 2];  // 36864 B
    __shared__ float sA[8][16 * SA_PITCH];          // 34816 B

    const int tid  = threadIdx.x;
    const int lane = tid & 31;
    const int wave = tid >> 5;
    const int lh   = lane & 15;
    const int half = lane >> 4;
    const int koff = 2 * half;                 // K sub-offset for this half

    const int tile = blockIdx.x * 8 + wave;
    const bool isDoc = (tile < DOC_TILES);
    const float* X   = isDoc ? dh : qh;
    const long rowbase = isDoc ? (long)tile * 16 : (long)(tile - DOC_TILES) * 16;

    v8f acc[8];
#pragma unroll
    for (int t = 0; t < 8; ++t) acc[t] = (v8f)0.0f;

    float* sAw = sA[wave];

    for (int chunk = 0; chunk < NCHUNK; ++chunk) {
        const int k0 = chunk * KC;
        __syncthreads();
        // --- stage W chunk pair-interleaved: sW[p][c] = {W[2p][c], W[2p+1][c]} ---
#pragma unroll
        for (int i = 0; i < 8; ++i) {
            int u  = tid + 256 * i;            // 2048 units, 2 (p,c) cells each
            int p  = u >> 6;                   // pair-row 0..31
            int c2 = (u & 63) * 2;             // col 0..126 step 2
            const float* src = &W[(long)(k0 + 2 * p) * CDIM + c2];
            v2f lo = *(const v2f*)src;               // W[2p][c2],  W[2p][c2+1]
            v2f hi = *(const v2f*)(src + CDIM);      // W[2p+1][c2],W[2p+1][c2+1]
            float4 v; v.x = lo.x; v.y = hi.x; v.z = lo.y; v.w = hi.y;
            *(float4*)&sW[2 * (p * SWP_PITCH + c2)] = v;
        }
        // --- stage this wave's 16 x 64 A strip (coalesced float4) ---
#pragma unroll
        for (int i = 0; i < 8; ++i) {
            int f   = lane + 32 * i;           // 256 float4 total
            int row = f >> 4;
            int c4  = (f & 15) * 4;
            float4 v = *(const float4*)&X[(rowbase + row) * HDIM + k0 + c4];
            *(float4*)&sAw[row * SA_PITCH + c4] = v;
        }
        __syncthreads();

        // --- 16 K-steps of V_WMMA_F32_16X16X4_F32 across 8 N-tiles ---
#pragma unroll 4
        for (int kk = 0; kk < KC; kk += 4) {
            v2f a = *(const v2f*)&sAw[lh * SA_PITCH + kk + koff];
            const int pbase = ((kk >> 1) + half) * SWP_PITCH;
            v2f bb[8];
#pragma unroll
            for (int t = 0; t < 8; ++t)
                bb[t] = *(const v2f*)&sW[2 * (pbase + 16 * t + lh)];
#pragma unroll
            for (int t = 0; t < 8; ++t)
                acc[t] = __builtin_amdgcn_wmma_f32_16x16x4_f32(
                    false, a, false, bb[t], (short)0, acc[t], false, false);
        }
    }

    // ------------- fused epilogue -------------
    float bias[8], wst[8];
    int cl[8];
#pragma unroll
    for (int t = 0; t < 8; ++t) {
        cl[t]  = 16 * t + lh;
        bias[t] = bcomp[cl[t]];
        wst[t]  = wstop[cl[t]];
    }
#pragma unroll
    for (int t = 0; t < 8; ++t)
#pragma unroll
        for (int j = 0; j < 8; ++j) acc[t][j] += bias[t];

    if (isDoc) {
        const float b0 = bstop[0];
        float imp[8];
#pragma unroll
        for (int j = 0; j < 8; ++j) {
            float p = 0.0f;
#pragma unroll
            for (int t = 0; t < 8; ++t) p += acc[t][j] * wst[t];
            // reduce across the 16 lanes of this half (cols of the row)
            p += __shfl_xor(p, 1);
            p += __shfl_xor(p, 2);
            p += __shfl_xor(p, 4);
            p += __shfl_xor(p, 8);
            p += b0;
            imp[j] = p > 0.0f ? p : 0.0f;
        }
#pragma unroll
        for (int j = 0; j < 8; ++j) {
            long row = rowbase + j + 8 * half;
            float sc = imp[j] * (float)dmask[row];
            float* dst = &dvecs[row * CDIM];
#pragma unroll
            for (int t = 0; t < 8; ++t) dst[cl[t]] = acc[t][j] * sc;
        }
    } else {
#pragma unroll
        for (int j = 0; j < 8; ++j) {
            long row = rowbase + j + 8 * half;
            float sc = (float)qmask[row];
            float* dst = &qvecs[row * CDIM];
#pragma unroll
            for (int t = 0; t < 8; ++t) dst[cl[t]] = acc[t][j] * sc;
        }
    }
}

__global__ __launch_bounds__(256) void colberter_score(
    const float* __restrict__ qh,     // (B*LQ, H) for cls row 0
    const float* __restrict__ dh,     // (B*LD, H) for cls row 0
    const int*   __restrict__ qmask,  // (B*LQ)
    const int*   __restrict__ dmask,  // (B*LD)
    const float* __restrict__ qvecs,  // (B*LQ, C)
    const float* __restrict__ dvecs,  // (B*LD, C)
    const float* __restrict__ merger, // (1)
    float* __restrict__ out)          // [score(128) | cls(128) | term(128)]
{
    __shared__ float s_term[4][32];
    __shared__ float s_cls[8];

    const int b    = blockIdx.x;
    const int tid  = threadIdx.x;
    const int lane = tid & 31;
    const int wave = tid >> 5;
    const int lh   = lane & 15;
    const int half = lane >> 4;
    const int koff = 2 * half;

    // ---- cls dot: q_cls . d_cls over H=768 ----
    {
        float cp = 0.0f;
        const float* qc = &qh[(long)b * LQ * HDIM];
        const float* dc = &dh[(long)b * LD * HDIM];
        for (int h = tid; h < HDIM; h += 256) cp += qc[h] * dc[h];
        cp += __shfl_xor(cp, 1);
        cp += __shfl_xor(cp, 2);
        cp += __shfl_xor(cp, 4);
        cp += __shfl_xor(cp, 8);
        cp += __shfl_xor(cp, 16);
        if (lane == 0) s_cls[wave] = cp;
    }

    // ---- scores = q_vecs (32xC) @ d_vecs^T (Cx512), tiled 16x16 ----
    const int m    = wave & 1;   // q row tile (0/1)
    const int ngrp = wave >> 1;  // doc tile group (0..3), 8 tiles each

    v8f acc[8];
#pragma unroll
    for (int i = 0; i < 8; ++i) acc[i] = (v8f)0.0f;

    const long qrow = (long)b * LQ + 16 * m + lh;
    const float* qrowp = &qvecs[qrow * CDIM];
    const float* dbase = &dvecs[((long)b * LD + ngrp * 8 * 16 + lh) * CDIM];

    for (int c = 0; c < CDIM; c += 4) {
        v2f a = *(const v2f*)&qrowp[c + koff];
        v2f bf[8];
#pragma unroll
        for (int i = 0; i < 8; ++i)
            bf[i] = *(const v2f*)&dbase[(long)i * 16 * CDIM + c + koff];
#pragma unroll
        for (int i = 0; i < 8; ++i)
            acc[i] = __builtin_amdgcn_wmma_f32_16x16x4_f32(
                false, a, false, bf[i], (short)0, acc[i], false, false);
    }

    // ---- masked max over this wave's 8 doc tiles ----
    float rm[8];
#pragma unroll
    for (int j = 0; j < 8; ++j) rm[j] = -1.0e30f;
#pragma unroll
    for (int i = 0; i < 8; ++i) {
        int dcol = (ngrp * 8 + i) * 16 + lh;          // doc index in batch
        bool dm = dmask[(long)b * LD + dcol] != 0;
#pragma unroll
        for (int j = 0; j < 8; ++j) {
            float v = dm ? acc[i][j] : -1000.0f;
            rm[j] = fmaxf(rm[j], v);
        }
    }
#pragma unroll
    for (int j = 0; j < 8; ++j) {
        rm[j] = fmaxf(rm[j], __shfl_xor(rm[j], 1));
        rm[j] = fmaxf(rm[j], __shfl_xor(rm[j], 2));
        rm[j] = fmaxf(rm[j], __shfl_xor(rm[j], 4));
        rm[j] = fmaxf(rm[j], __shfl_xor(rm[j], 8));
    }
    if (lh == 0) {
#pragma unroll
        for (int j = 0; j < 8; ++j)
            s_term[ngrp][16 * m + j + 8 * half] = rm[j];
    }
    __syncthreads();

    // ---- combine groups, masked sum over q rows, merge ----
    if (wave == 0) {
        int r = lane;
        float term = fmaxf(fmaxf(s_term[0][r], s_term[1][r]),
                           fmaxf(s_term[2][r], s_term[3][r]));
        float val = (qmask[(long)b * LQ + r] != 0) ? term : 0.0f;
        val += __shfl_xor(val, 1);
        val += __shfl_xor(val, 2);
        val += __shfl_xor(val, 4);
        val += __shfl_xor(val, 8);
        val += __shfl_xor(val, 16);
        if (lane == 0) {
            float cls = 0.0f;
#pragma unroll
            for (int i = 0; i < 8; ++i) cls += s_cls[i];
            float w = 1.0f / (1.0f + __expf(-merger[0]));
            float clsS  = cls * w;
            float termS = val * (1.0f - w);
            out[b]       = clsS + termS;
            out[128 + b] = clsS;
            out[256 + b] = termS;
        }
    }
}

extern "C" void kernel_launch(void* const* d_in, const int* in_sizes, int n_in,
                              void* d_out, int out_size, void* d_ws, size_t ws_size,
                              hipStream_t stream) {
    (void)in_sizes; (void)n_in; (void)out_size; (void)ws_size;
    const float* qh     = (const float*)d_in[0];
    const float* dh     = (const float*)d_in[1];
    const int*   qmask  = (const int*)d_in[2];
    const int*   dmask  = (const int*)d_in[3];
    const float* W      = (const float*)d_in[4];
    const float* bcomp  = (const float*)d_in[5];
    const float* wstop  = (const float*)d_in[6];
    const float* bstop  = (const float*)d_in[7];
    const float* merger = (const float*)d_in[8];
    float* out = (float*)d_out;

    float* qvecs = (float*)d_ws;                          // 4096*128 f32 = 2 MB
    float* dvecs = qvecs + (size_t)QRY_ROWS * CDIM;       // 65536*128 f32 = 32 MB

    colberter_proj<<<ALL_TILES / 8, 256, 0, stream>>>(
        qh, dh, qmask, dmask, W, bcomp, wstop, bstop, qvecs, dvecs);
    colberter_score<<<BCOUNT, 256, 0, stream>>>(
        qh, dh, qmask, dmask, qvecs, dvecs, merger, out);
}